// DynamicKMatcher_10316511445743
// MI455X (gfx1250) — compile-verified
//
#include <hip/hip_runtime.h>
#include <stdint.h>

// DynamicK (SimOTA-style) matcher for gfx1250 / MI455X.
// Fixed problem shape from setup_inputs(): B=8, N=8000, G=250, OTA_K=5.
// Workspace use: 150000 floats (600 KB) — assumed ws_size >= 600 KB.
// Output: [mask (B*N) as 0/1 float][gidx (B*N) as float], out_size = 128000.
//
// No WMMA: the pairwise cost (abs/min/max/clip/div) has no K-contraction, so
// matrix cores don't apply. CDNA5-specific paths used instead:
//   * global_load_async_to_lds_b128 + s_wait_asynccnt (double-buffered tiles)
//   * wave32 shuffle butterflies for top-k merge
//   * LDS 64-bit atomicMin (ds_min_u64) for conflict resolution
//   * global_prefetch_b8 in the streaming column argmin

#define NB 8
#define NP 8000
#define NG 250
#define KK 5
#define TILE 800
#define NTILE (NP / TILE)

#define ALPHA_F 0.25f
#define EPS_F 1e-8f
#define CR_F 2.5f
#define W_CLASS_F 2.0f
#define W_L1_F 5.0f
#define W_GIOU_F 2.0f

// ---------- helpers ----------
__device__ __forceinline__ uint32_t ford(float f) {
  uint32_t u = __float_as_uint(f);
  return (u & 0x80000000u) ? ~u : (u | 0x80000000u);
}
__device__ __forceinline__ float finv(uint32_t o) {
  uint32_t u = (o & 0x80000000u) ? (o ^ 0x80000000u) : ~o;
  return __uint_as_float(u);
}
__device__ __forceinline__ uint32_t lds_off(const void* p) {
  // gfx1250: flat LDS aperture keeps the LDS byte offset in addr[31:0].
  return (uint32_t)(uintptr_t)p;
}
__device__ __forceinline__ void async_b128(uint32_t lds, uint32_t voff, const void* base) {
  asm volatile("global_load_async_to_lds_b128 %0, %1, %2"
               :: "v"(lds), "v"(voff),
                  "s"((unsigned long long)(uintptr_t)base)
               : "memory");
}
__device__ __forceinline__ void wait_async0() {
  asm volatile("s_wait_asynccnt 0" ::: "memory");
}
__device__ __forceinline__ uint64_t shfl_xor_u64(uint64_t v, int m) {
  uint32_t lo = (uint32_t)v, hi = (uint32_t)(v >> 32);
  lo = __shfl_xor(lo, m, 32);
  hi = __shfl_xor(hi, m, 32);
  return ((uint64_t)hi << 32) | lo;
}
__device__ __forceinline__ uint64_t wave_min_u64(uint64_t v) {
  for (int off = 16; off > 0; off >>= 1) {
    uint64_t o = shfl_xor_u64(v, off);
    v = (o < v) ? o : v;
  }
  return v;
}
__device__ __forceinline__ uint64_t wave_max_u64(uint64_t v) {
  for (int off = 16; off > 0; off >>= 1) {
    uint64_t o = shfl_xor_u64(v, off);
    v = (o > v) ? o : v;
  }
  return v;
}

struct GtPre {
  float gx0, gy0, gx1, gy1;  // gt xyxy
  float n0, n1, n2, n3;      // normalized xyxy
  float ag;                  // gt area
};
__device__ __forceinline__ GtPre gt_pre(float cx, float cy, float w, float h,
                                        float w0, float w1, float w2, float w3) {
  GtPre r;
  r.gx0 = cx - w * 0.5f; r.gy0 = cy - h * 0.5f;
  r.gx1 = cx + w * 0.5f; r.gy1 = cy + h * 0.5f;
  r.n0 = r.gx0 / w0; r.n1 = r.gy0 / w1; r.n2 = r.gx1 / w2; r.n3 = r.gy1 / w3;
  r.ag = (r.gx1 - r.gx0) * (r.gy1 - r.gy0);
  return r;
}

__device__ __forceinline__ float pair_cost(float px0, float py0, float px1, float py1,
                                           const GtPre& G, float ccls, float lo,
                                           float w0, float w1, float w2, float w3,
                                           float* iouOut) {
  float pcx = (px0 + px1) * 0.5f, pcy = (py0 + py1) * 0.5f;
  bool strict = (pcx > G.gx0) && (pcx < G.gx1) && (pcy > G.gy0) && (pcy < G.gy1);
  float ap = (px1 - px0) * (py1 - py0);
  float ix0 = fmaxf(px0, G.gx0), iy0 = fmaxf(py0, G.gy0);
  float ix1 = fminf(px1, G.gx1), iy1 = fminf(py1, G.gy1);
  float iw = fmaxf(ix1 - ix0, 0.0f), ih = fmaxf(iy1 - iy0, 0.0f);
  float inter = iw * ih;
  float uni = ap + G.ag - inter;
  float iou = inter / uni;
  float hx0 = fminf(px0, G.gx0), hy0 = fminf(py0, G.gy0);
  float hx1 = fmaxf(px1, G.gx1), hy1 = fmaxf(py1, G.gy1);
  float hw = fmaxf(hx1 - hx0, 0.0f), hh = fmaxf(hy1 - hy0, 0.0f);
  float ah = hw * hh;
  float giou = iou - (ah - uni) / ah;
  float cb = fabsf(px0 / w0 - G.n0) + fabsf(py0 / w1 - G.n1) +
             fabsf(px1 / w2 - G.n2) + fabsf(py1 / w3 - G.n3);
  float c = W_L1_F * cb + W_CLASS_F * ccls + W_GIOU_F * (-giou);
  c += strict ? 0.0f : 100.0f;
  c += (lo != 0.0f) ? 0.0f : 10000.0f;
  *iouOut = iou;
  return c;
}

// ---------- kernel 1: per-pred cost_class + loose flag ----------
__global__ __launch_bounds__(256) void prep_kernel(
    const float* __restrict__ logits, const float* __restrict__ pxyxy,
    const float* __restrict__ gt, float* __restrict__ cost_class,
    float* __restrict__ loose) {
  __shared__ __attribute__((aligned(16))) float s_gt[NG * 4];
  const int b = blockIdx.y;
  const int tid = threadIdx.x;
  const float* gtb = gt + (size_t)b * NG * 4;
  if (tid < NG) async_b128(lds_off(&s_gt[tid * 4]), (uint32_t)(tid * 16), gtb);
  wait_async0();
  __syncthreads();
  const int n = blockIdx.x * 256 + tid;
  if (n >= NP) return;
  float l = logits[(size_t)b * NP + n];
  float p = 1.0f / (1.0f + expf(-l));
  float pos = ALPHA_F * (1.0f - p) * (1.0f - p) * (-logf(p + EPS_F));
  float neg = (1.0f - ALPHA_F) * p * p * (-logf(1.0f - p + EPS_F));
  cost_class[(size_t)b * NP + n] = pos - neg;
  float4 pb4 = ((const float4*)pxyxy)[(size_t)b * NP + n];
  float pcx = (pb4.x + pb4.z) * 0.5f, pcy = (pb4.y + pb4.w) * 0.5f;
  bool any = false;
  for (int g = 0; g < NG; ++g) {
    float4 gb = ((const float4*)s_gt)[g];
    float gx0 = gb.x - gb.z * 0.5f, gx1 = gb.x + gb.z * 0.5f;
    float gy0 = gb.y - gb.w * 0.5f, gy1 = gb.y + gb.w * 0.5f;
    float gw = gx1 - gx0, gh = gy1 - gy0;
    bool c = (pcx > gb.x - CR_F * gw) && (pcx < gb.x + CR_F * gw) &&
             (pcy > gb.y - CR_F * gh) && (pcy < gb.y + CR_F * gh);
    any = any || c;
  }
  loose[(size_t)b * NP + n] = any ? 1.0f : 0.0f;
}

// ---------- kernel 2: per-column streaming top-k (one wave per gt) ----------
__global__ __launch_bounds__(256) void cost_topk_kernel(
    const float* __restrict__ pxyxy, const float* __restrict__ gt,
    const float* __restrict__ cost_class, const float* __restrict__ loose,
    const float* __restrict__ whwh,
    int* __restrict__ cand_idx, float* __restrict__ cand_cost,
    int* __restrict__ cand_k) {
  __shared__ __attribute__((aligned(16))) float s_gt[NG * 4];
  __shared__ __attribute__((aligned(16))) float s_box[2][TILE * 4];
  const int b = blockIdx.y;
  const int tid = threadIdx.x;
  const int wv = tid >> 5, lane = tid & 31;
  const int g = blockIdx.x * 8 + wv;
  const float* gtb = gt + (size_t)b * NG * 4;
  const float* pb = pxyxy + (size_t)b * NP * 4;
  const float* ccb = cost_class + (size_t)b * NP;
  const float* lob = loose + (size_t)b * NP;
  const float w0 = whwh[0], w1 = whwh[1], w2 = whwh[2], w3 = whwh[3];

  // Prologue: async-stage gts + pred tile 0 into LDS.
  if (tid < NG) async_b128(lds_off(&s_gt[tid * 4]), (uint32_t)(tid * 16), gtb);
  for (int j = tid; j < TILE; j += 256)
    async_b128(lds_off(&s_box[0][j * 4]), (uint32_t)(j * 16), pb);
  wait_async0();
  __syncthreads();

  const bool active = (g < NG);
  GtPre GP = {};
  if (active) {
    float4 gb = ((const float4*)s_gt)[g];
    GP = gt_pre(gb.x, gb.y, gb.z, gb.w, w0, w1, w2, w3);
  }
  uint64_t ck[KK], ik[KK];
#pragma unroll
  for (int r = 0; r < KK; ++r) { ck[r] = ~0ull; ik[r] = 0ull; }

  for (int t = 0; t < NTILE; ++t) {
    const int buf = t & 1;
    if (t + 1 < NTILE) {  // async-prefetch next tile while computing this one
      const int base = (t + 1) * TILE;
      for (int j = tid; j < TILE; j += 256)
        async_b128(lds_off(&s_box[buf ^ 1][j * 4]), (uint32_t)((base + j) * 16), pb);
    }
    if (active) {
      for (int j = lane; j < TILE; j += 32) {
        const int n = t * TILE + j;
        float4 p4 = ((const float4*)s_box[buf])[j];
        float iou;
        float c = pair_cost(p4.x, p4.y, p4.z, p4.w, GP, ccb[n], lob[n],
                            w0, w1, w2, w3, &iou);
        uint64_t kc = ((uint64_t)ford(c) << 32) | (uint32_t)n;
        if (kc < ck[KK - 1]) {  // insert ascending (stable-argsort tie order)
          int r = KK - 1;
          while (r > 0 && ck[r - 1] > kc) { ck[r] = ck[r - 1]; --r; }
          ck[r] = kc;
        }
        uint64_t ki = ((uint64_t)ford(iou) << 32) | (uint32_t)n;
        if (ki > ik[KK - 1]) {  // insert descending
          int r = KK - 1;
          while (r > 0 && ik[r - 1] < ki) { ik[r] = ik[r - 1]; --r; }
          ik[r] = ki;
        }
      }
    }
    wait_async0();
    __syncthreads();
  }
  if (!active) return;

  // Wave32 merge: 5 pop-rounds of min (cost) / max (iou). Keys are unique.
  uint64_t csel[KK];
  int cc = 0;
  for (int r = 0; r < KK; ++r) {
    uint64_t v = (cc < KK) ? ck[cc] : ~0ull;
    uint64_t m = wave_min_u64(v);
    if (cc < KK && ck[cc] == m) ++cc;
    csel[r] = m;
  }
  float isum = 0.0f;
  int ci = 0;
  for (int r = 0; r < KK; ++r) {
    uint64_t v = (ci < KK) ? ik[ci] : 0ull;
    uint64_t m = wave_max_u64(v);
    if (ci < KK && ik[ci] == m) ++ci;
    isum += finv((uint32_t)(m >> 32));
  }
  if (lane == 0) {
    int k = (int)isum;            // trunc like astype(int32)
    if (k < 1) k = 1;
    if (k > KK) k = KK;
    const int base = (b * NG + g) * KK;
    for (int r = 0; r < KK; ++r) {
      cand_idx[base + r] = (int)(csel[r] & 0xffffffffu);
      cand_cost[base + r] = finv((uint32_t)(csel[r] >> 32));
    }
    cand_k[b * NG + g] = k;
  }
}

// ---------- kernel 3: per-batch conflict resolution + while-loop ----------
__global__ __launch_bounds__(256) void match_kernel(
    const float* __restrict__ pxyxy, const float* __restrict__ gt,
    const float* __restrict__ cost_class, const float* __restrict__ loose,
    const float* __restrict__ whwh,
    const int* __restrict__ cand_idx, const float* __restrict__ cand_cost,
    const int* __restrict__ cand_k, float* __restrict__ out) {
  __shared__ __attribute__((aligned(16))) float s_gt[NG * 4];
  __shared__ unsigned long long bk[NP];   // per-pred min (cost<<32)|g
  __shared__ int asg[NP];                 // assigned gt or -1
  __shared__ unsigned char infRow[NP];    // rows inf'd in cc (ever matched)
  __shared__ int cnt[NG];
  __shared__ int ugList[NG];
  __shared__ int posArr[NG];
  __shared__ float pcostArr[NG];
  __shared__ int ugCnt;
  const int b = blockIdx.x, tid = threadIdx.x, wv = tid >> 5, lane = tid & 31;
  const float* gtb = gt + (size_t)b * NG * 4;
  const float* pb = pxyxy + (size_t)b * NP * 4;
  const float* ccb = cost_class + (size_t)b * NP;
  const float* lob = loose + (size_t)b * NP;
  const float w0 = whwh[0], w1 = whwh[1], w2 = whwh[2], w3 = whwh[3];

  if (tid < NG) async_b128(lds_off(&s_gt[tid * 4]), (uint32_t)(tid * 16), gtb);
  for (int n = tid; n < NP; n += 256) { bk[n] = ~0ull; asg[n] = -1; infRow[n] = 0; }
  if (tid < NG) cnt[tid] = 0;
  wait_async0();
  __syncthreads();

  // Initial matches: top-k candidates + conflict resolution (min over (cost,g)).
  const int cbase = b * NG * KK;
  for (int e = tid; e < NG * KK; e += 256) {
    int g2 = e / KK, j = e % KK;
    if (j < cand_k[b * NG + g2]) {
      int n = cand_idx[cbase + e];
      uint64_t key = ((uint64_t)ford(cand_cost[cbase + e]) << 32) | (uint32_t)g2;
      atomicMin(&bk[n], (unsigned long long)key);
    }
  }
  __syncthreads();
  for (int e = tid; e < NG * KK; e += 256) {
    int g2 = e / KK, j = e % KK;
    if (j < cand_k[b * NG + g2]) {
      int n = cand_idx[cbase + e];
      uint64_t key = ((uint64_t)ford(cand_cost[cbase + e]) << 32) | (uint32_t)g2;
      if (bk[n] == (unsigned long long)key) { asg[n] = g2; atomicAdd(&cnt[g2], 1); }
    }
  }
  __syncthreads();

  int iter = 0;
  while (true) {
    if (tid == 0) ugCnt = 0;
    __syncthreads();
    if (tid < NG && cnt[tid] == 0) { int i = atomicAdd(&ugCnt, 1); ugList[i] = tid; }
    __syncthreads();
    const int nu = ugCnt;
    if (nu == 0 || iter >= NP) break;
    for (int n = tid; n < NP; n += 256)
      if (asg[n] >= 0) infRow[n] = 1;   // cc rows of matched preds -> inf (cumulative)
    __syncthreads();
    // One wave per unmatched gt: argmin over recomputed column.
    for (int u = wv; u < nu; u += 8) {
      const int g2 = ugList[u];
      float4 gb = ((const float4*)s_gt)[g2];
      GtPre GP = gt_pre(gb.x, gb.y, gb.z, gb.w, w0, w1, w2, w3);
      uint64_t best = ~0ull;
      for (int n = lane; n < NP; n += 32) {
        int pf = n + 128; if (pf >= NP) pf = NP - 1;
        __builtin_prefetch(&pb[(size_t)pf * 4], 0, 0);
        float c;
        if (infRow[n]) {
          c = __builtin_inff();
        } else {
          float4 p4 = ((const float4*)pb)[n];
          float iou;
          c = pair_cost(p4.x, p4.y, p4.z, p4.w, GP, ccb[n], lob[n],
                        w0, w1, w2, w3, &iou);
        }
        uint64_t k2 = ((uint64_t)ford(c) << 32) | (uint32_t)n;
        best = (k2 < best) ? k2 : best;
      }
      best = wave_min_u64(best);
      if (lane == 0) {
        int p = (int)(best & 0xffffffffu);
        posArr[u] = p;
        pcostArr[u] = finv((uint32_t)(best >> 32));
        bk[p] = ~0ull;  // reset stale resolution key for this pred
      }
    }
    __syncthreads();
    // Resolve conflicts among the new adds only (same semantics as reference:
    // a newly added pred keeps the (cost,g)-min column; losers stay unmatched).
    for (int u = tid; u < nu; u += 256) {
      const int g2 = ugList[u], p = posArr[u];
      const float pc = pcostArr[u];
      if (pc < __builtin_inff())
        atomicMin(&bk[p], (unsigned long long)(((uint64_t)ford(pc) << 32) | (uint32_t)g2));
    }
    __syncthreads();
    for (int u = tid; u < nu; u += 256) {
      const int g2 = ugList[u], p = posArr[u];
      const float pc = pcostArr[u];
      uint64_t key = ((uint64_t)ford(pc) << 32) | (uint32_t)g2;
      if (pc < __builtin_inff() && bk[p] == (unsigned long long)key && asg[p] < 0) {
        asg[p] = g2;
        atomicAdd(&cnt[g2], 1);
      }
    }
    __syncthreads();
    ++iter;
  }

  for (int n = tid; n < NP; n += 256) {
    const int a = asg[n];
    out[(size_t)b * NP + n] = (a >= 0) ? 1.0f : 0.0f;                 // mask
    out[(size_t)NB * NP + (size_t)b * NP + n] = (float)a;             // gidx (-1 if none)
  }
}

// ---------- launch ----------
extern "C" void kernel_launch(void* const* d_in, const int* in_sizes, int n_in,
                              void* d_out, int out_size, void* d_ws, size_t ws_size,
                              hipStream_t stream) {
  const float* logits = (const float*)d_in[0];  // (B,N,1)
  const float* pxyxy  = (const float*)d_in[1];  // (B,N,4)
  const float* gt     = (const float*)d_in[2];  // (B,G,4)
  const float* whwh   = (const float*)d_in[3];  // (4,)
  float* out = (float*)d_out;

  float* ws = (float*)d_ws;
  float* cost_class = ws;                                   // NB*NP
  float* loose      = ws + (size_t)NB * NP;                 // NB*NP
  float* cand_cost  = ws + (size_t)2 * NB * NP;             // NB*NG*KK
  int*   cand_idx   = (int*)(ws + (size_t)2 * NB * NP + (size_t)NB * NG * KK);
  int*   cand_k     = (int*)(ws + (size_t)2 * NB * NP + (size_t)2 * NB * NG * KK);

  (void)in_sizes; (void)n_in; (void)out_size; (void)ws_size;

  prep_kernel<<<dim3((NP + 255) / 256, NB), 256, 0, stream>>>(
      logits, pxyxy, gt, cost_class, loose);
  cost_topk_kernel<<<dim3((NG + 7) / 8, NB), 256, 0, stream>>>(
      pxyxy, gt, cost_class, loose, whwh, cand_idx, cand_cost, cand_k);
  match_kernel<<<NB, 256, 0, stream>>>(
      pxyxy, gt, cost_class, loose, whwh, cand_idx, cand_cost, cand_k, out);
}